// Encoder_z0_signature_55611236549352
// MI455X (gfx1250) — compile-verified
//
#include <hip/hip_runtime.h>
#include <hip/hip_bf16.h>
#include <math.h>

typedef __attribute__((ext_vector_type(2))) float        v2f;
typedef __attribute__((ext_vector_type(8))) float        v8f;
typedef __attribute__((ext_vector_type(4))) unsigned int v4u;
typedef __attribute__((ext_vector_type(4))) int          v4i;
typedef __attribute__((ext_vector_type(8))) int          v8i;

// Problem constants (from reference)
#define BATCH   2048
#define LSEQ    70
#define DIN     15
#define HID     16
#define NF      4
#define KW      40
#define LP      31            // LSEQ - KW + 1
#define CDIM    20            // DIN + 1 + NF
#define SIGCH   8420          // CDIM + CDIM^2 + CDIM^3
#define SIGP    8448          // padded: multiple of KC -> branch-free GEMM
#define Z0      64
#define NOUT    128           // 2*Z0

#define KC      32            // K-chunk per LDS stage (32 DW -> TDM pad interval 4)
#define PADS    33            // 32+1 stride (odd) -> conflict-free column reads
#define NCH     (SIGP / KC)   // 264

#if defined(__has_builtin)
#  if __has_builtin(__builtin_amdgcn_tensor_load_to_lds)
#    define HAVE_TDM 1
#  endif
#endif

// ---------------------------------------------------------------------------
// TDM helper: async-load a 2D tile (tile_d1 rows x KC f32 cols, row stride
// SIGP elements) from global into LDS, inserting 1 pad DWORD after every
// 32 DWORDs (pad_interval code 4) so the LDS image has stride PADS=33.
// Descriptor packing per CDNA5 ISA.
// ---------------------------------------------------------------------------
#ifdef HAVE_TDM
__device__ __forceinline__ void tdm_load_tile(unsigned lds_addr,
                                              const float* gptr,
                                              int tile_d1)
{
    const unsigned long long ga = (unsigned long long)(uintptr_t)gptr;
    v4u g0;
    g0.x = 1u;                                    // count=1, user mode
    g0.y = lds_addr;                              // lds_addr [63:32]
    g0.z = (unsigned)(ga & 0xFFFFFFFFu);          // global_addr [95:64]
    g0.w = (unsigned)((ga >> 32) & 0x01FFFFFFu)   // global_addr [120:96]
         | (2u << 30);                            // type=2 ("image")
    v8i g1;
    g1[0] = (int)((2u << 16)        // data_size = 4 bytes
                | (1u << 20)        // pad_enable
                | (4u << 22));      // pad_interval code 4 = 32 DWORDs; amount=1 DW
    g1[1] = (int)(((unsigned)SIGP & 0xFFFFu) << 16);       // tensor_dim0 lo16
    g1[2] = (int)(((unsigned)SIGP >> 16)                   // tensor_dim0 hi16
                | (((unsigned)tile_d1 & 0xFFFFu) << 16));  // tensor_dim1 lo16
    g1[3] = (int)(((unsigned)KC & 0xFFFFu) << 16);         // tile_dim0 = KC
    g1[4] = (int)((unsigned)tile_d1 & 0xFFFFu);            // tile_dim1
    g1[5] = (int)SIGP;                                     // tensor_dim0_stride lo32
    g1[6] = 0;
    g1[7] = 0;
    const v4i z4 = {0, 0, 0, 0};
#if __clang_major__ >= 23
    const v8i z8 = {0, 0, 0, 0, 0, 0, 0, 0};
    __builtin_amdgcn_tensor_load_to_lds(g0, g1, z4, z4, z8, 0);
#else
    __builtin_amdgcn_tensor_load_to_lds(g0, g1, z4, z4, 0);
#endif
}
__device__ __forceinline__ unsigned lds_off(const void* p) {
    return (unsigned)(unsigned long long)(uintptr_t)p;  // LDS aperture: low 32b
}
#endif

// ---------------------------------------------------------------------------
// Kernel W: copy w_out (128 x 8420) into zero-padded (128 x 8448) image.
// ---------------------------------------------------------------------------
__global__ __launch_bounds__(256)
void pad_wout_kernel(const float* __restrict__ w_out, float* __restrict__ w_pad)
{
    const int idx = blockIdx.x * 256 + threadIdx.x;      // < NOUT*SIGP
    const int row = idx / SIGP;
    const int c   = idx - row * SIGP;
    if (row < NOUT)
        w_pad[idx] = (c < SIGCH) ? w_out[row * SIGCH + c] : 0.0f;
}

// ---------------------------------------------------------------------------
// Kernel A: one block (512 threads) per batch element.
//   conv1 -> relu -> 16x16 mm -> relu -> 4x16 mm -> path(31x20) in LDS,
//   then depth-3 signature scan: thread (i,j) owns s2 scalar and the
//   s3[i][j][0..19] row in registers. Emits padded sig row (8448).
// ---------------------------------------------------------------------------
__global__ __launch_bounds__(512)
void sig_path_kernel(const float* __restrict__ x,      // (B, L, DIN)
                     const float* __restrict__ w1,     // (HID, DIN, KW)
                     const float* __restrict__ b1,     // (HID)
                     const float* __restrict__ w2,     // (HID, HID)
                     const float* __restrict__ b2,     // (HID)
                     const float* __restrict__ w3,     // (NF, HID)
                     const float* __restrict__ b3,     // (NF)
                     float* __restrict__ sig)          // (B, SIGP)
{
    __shared__ float x_s[LSEQ * DIN];          // 1050
    __shared__ float w1_s[HID * DIN * KW];     // 9600
    __shared__ float h1_s[HID * LP];           // 496 (post-relu)
    __shared__ float h2_s[HID * LP];           // 496 (post-relu)
    __shared__ float path_s[LP * CDIM];        // 620
    __shared__ float incs_s[LP * CDIM];        // 620

    const int b   = blockIdx.x;
    const int tid = threadIdx.x;

    // ---- stage x and conv1 weights ----
    for (int i = tid; i < LSEQ * DIN; i += 512) x_s[i]  = x[b * (LSEQ * DIN) + i];
    for (int i = tid; i < HID * DIN * KW; i += 512) w1_s[i] = w1[i];
    __syncthreads();

    // ---- conv1 + relu ----
    if (tid < HID * LP) {
        const int o = tid / LP, l = tid % LP;
        float acc = b1[o];
        const float* wrow = &w1_s[o * (DIN * KW)];
        for (int c = 0; c < DIN; ++c) {
            #pragma unroll 8
            for (int k = 0; k < KW; ++k)
                acc = fmaf(x_s[(l + k) * DIN + c], wrow[c * KW + k], acc);
        }
        h1_s[o * LP + l] = fmaxf(acc, 0.0f);
    }
    __syncthreads();

    // ---- 16x16 matmul + relu ----
    if (tid < HID * LP) {
        const int o = tid / LP, l = tid % LP;
        float acc = b2[o];
        #pragma unroll
        for (int c = 0; c < HID; ++c)
            acc = fmaf(w2[o * HID + c], h1_s[c * LP + l], acc);
        h2_s[o * LP + l] = fmaxf(acc, 0.0f);
    }
    __syncthreads();

    // ---- 4x16 matmul + concat into path ----
    if (tid < NF * LP) {
        const int f = tid / LP, l = tid % LP;
        float acc = b3[f];
        #pragma unroll
        for (int c = 0; c < HID; ++c)
            acc = fmaf(w3[f * HID + c], h2_s[c * LP + l], acc);
        path_s[l * CDIM + (DIN + 1 + f)] = acc;
    }
    if (tid >= 128 && tid < 128 + LP * (DIN + 1)) {   // raw window + time
        const int t2 = tid - 128;
        const int l = t2 / (DIN + 1), ch = t2 % (DIN + 1);
        path_s[l * CDIM + ch] =
            (ch < DIN) ? x_s[(l + KW - 1) * DIN + ch]
                       : (float)l * (1.0f / (float)(LP - 1));
    }
    __syncthreads();

    // ---- increments (prepend-zero diff) ----
    if (tid < LP * CDIM) {
        const int l = tid / CDIM;
        incs_s[tid] = path_s[tid] - (l > 0 ? path_s[tid - CDIM] : 0.0f);
    }
    __syncthreads();

    // ---- depth-3 signature scan + emit ----
    float* outr = &sig[(size_t)b * SIGP];
    if (tid < CDIM * CDIM) {
        const int i = tid / CDIM, j = tid % CDIM;
        float s1 = 0.0f, s2 = 0.0f;
        float s3[CDIM];
        #pragma unroll
        for (int k = 0; k < CDIM; ++k) s3[k] = 0.0f;

        for (int t = 0; t < LP; ++t) {
            const float* d = &incs_s[t * CDIM];
            const float di = d[i];
            const float dj = d[j];
            // n3[i][j][k] = s3 + d[k]*(s2 + .5*dj*(s1 + di/3))
            const float coef = s2 + 0.5f * dj * (s1 + di * (1.0f / 3.0f));
            #pragma unroll
            for (int k = 0; k < CDIM; ++k)
                s3[k] = fmaf(coef, d[k], s3[k]);
            s2 = fmaf(s1, dj, fmaf(0.5f * di, dj, s2));   // uses OLD s1
            s1 += di;
        }

        if (j == 0) outr[i] = s1;                          // level-1 (20)
        outr[CDIM + i * CDIM + j] = s2;                    // level-2 (400)
        float* r3 = &outr[CDIM + CDIM * CDIM + (i * CDIM + j) * CDIM];
        #pragma unroll
        for (int k = 0; k < CDIM; ++k) r3[k] = s3[k];      // level-3 (8000)
    } else if (tid >= 400 && tid < 400 + (SIGP - SIGCH)) {
        outr[SIGCH + (tid - 400)] = 0.0f;                  // K padding
    }
}

// ---------------------------------------------------------------------------
// Kernel B: z = sig @ w_out^T + b_out via V_WMMA_F32_16X16X4_F32.
// Block = 8 waves = one 16-row M-strip x all 128 cols. Per 32-wide K-chunk:
// TDM double-buffered staging (wave0 -> B panel 128x32, wave1 -> A tile
// 16x32; both land in LDS with pad-33 stride straight from the DMA), while
// all waves run software-pipelined WMMA on the other buffer.
// ---------------------------------------------------------------------------
__global__ __launch_bounds__(256)
void sig_gemm_out_kernel(const float* __restrict__ sig,    // (B, SIGP)
                         const float* __restrict__ w_pad,  // (NOUT, SIGP)
                         const float* __restrict__ b_out,  // (NOUT)
                         float* __restrict__ out)          // mean|std flat
{
    __shared__ float a_tile[2][16 * PADS];        //  4.2 KB
    __shared__ float b_tile[2][NOUT * PADS];      // 33.8 KB

    const int wave = threadIdx.x >> 5;       // 0..7  -> N tile
    const int lane = threadIdx.x & 31;
    const int r0   = blockIdx.x * 16;        // M rows of this block
    const int col0 = wave * 16;              // N cols of this wave

    const int m  = lane & 15;                // A row / B col within tile
    const int kb = (lane >> 4) * 2;          // K sub-offset (ISA 16x4 layout)

    v8f acc = {};

#ifdef HAVE_TDM
    // prologue: DMA chunk 0 into buffer 0
    if (wave == 0)      tdm_load_tile(lds_off(&b_tile[0][0]), w_pad, NOUT);
    else if (wave == 1) tdm_load_tile(lds_off(&a_tile[0][0]),
                                      sig + (size_t)r0 * SIGP, 16);
#endif

    for (int ch = 0; ch < NCH; ++ch) {
        const int cur = ch & 1;
        const int nxt = cur ^ 1;
        const int kcn = (ch + 1) * KC;

#ifdef HAVE_TDM
        if (ch + 1 < NCH) {
            if (wave == 0)
                tdm_load_tile(lds_off(&b_tile[nxt][0]), w_pad + kcn, NOUT);
            else if (wave == 1)
                tdm_load_tile(lds_off(&a_tile[nxt][0]),
                              sig + (size_t)r0 * SIGP + kcn, 16);
            if (wave < 2) __builtin_amdgcn_s_wait_tensorcnt(1);
        } else {
            if (wave < 2) __builtin_amdgcn_s_wait_tensorcnt(0);
        }
#else
        // fallback: cooperative staging of chunk `ch` into buffer `cur`
        const int kc = ch * KC;
        for (int idx = threadIdx.x; idx < 16 * KC; idx += 256) {
            const int am = idx / KC, ak = idx % KC;
            a_tile[cur][am * PADS + ak] = sig[(size_t)(r0 + am) * SIGP + kc + ak];
        }
        #pragma unroll
        for (int rep = 0; rep < (NOUT * KC) / 256; ++rep) {
            const int idx = rep * 256 + threadIdx.x;
            const int bc = idx / KC, ak = idx % KC;
            b_tile[cur][bc * PADS + ak] = w_pad[(size_t)bc * SIGP + kc + ak];
        }
        (void)nxt; (void)kcn;
#endif
        __syncthreads();

        const float* at = &a_tile[cur][m * PADS + kb];
        const float* bt = &b_tile[cur][(col0 + m) * PADS + kb];

        // software-pipelined WMMA: fragments for step ks+4 are fetched
        // before the WMMA of step ks, hiding LDS latency.
        v2f a0, b0;
        a0.x = at[0]; a0.y = at[1];
        b0.x = bt[0]; b0.y = bt[1];
        #pragma unroll
        for (int ks = 0; ks < KC; ks += 4) {
            const int kn = (ks + 4 < KC) ? ks + 4 : ks;
            v2f a1, b1;
            a1.x = at[kn];     a1.y = at[kn + 1];
            b1.x = bt[kn];     b1.y = bt[kn + 1];
            acc = __builtin_amdgcn_wmma_f32_16x16x4_f32(
                      false, a0, false, b0, (short)0, acc, false, false);
            a0 = a1; b0 = b1;
        }
        __syncthreads();
    }

    // D layout: element (row v + 8*(lane>=16), col lane&15) lives in VGPR v
    const int n_g  = col0 + (lane & 15);
    const int rowh = (lane >> 4) * 8;
    const float bias = b_out[n_g];
    #pragma unroll
    for (int v = 0; v < 8; ++v) {
        const int row = r0 + rowh + v;
        const float z = acc[v] + bias;
        if (n_g < Z0) {
            out[(size_t)row * Z0 + n_g] = z;                       // mean_z0
        } else {
            const float sp = (z > 20.0f) ? z : log1pf(__expf(z));  // softplus
            out[(size_t)BATCH * Z0 + (size_t)row * Z0 + (n_g - Z0)] = sp;
        }
    }
}

// ---------------------------------------------------------------------------
extern "C" void kernel_launch(void* const* d_in, const int* in_sizes, int n_in,
                              void* d_out, int out_size, void* d_ws, size_t ws_size,
                              hipStream_t stream) {
    const float* x     = (const float*)d_in[0];  // observed_data (B,L,DIN)
    // d_in[1] = observed_tp -- unused by reference (it rebuilds linspace)
    const float* w1    = (const float*)d_in[2];
    const float* b1    = (const float*)d_in[3];
    const float* w2    = (const float*)d_in[4];
    const float* b2    = (const float*)d_in[5];
    const float* w3    = (const float*)d_in[6];
    const float* b3    = (const float*)d_in[7];
    const float* w_out = (const float*)d_in[8];
    const float* b_out = (const float*)d_in[9];
    float* out = (float*)d_out;

    float* sig   = (float*)d_ws;                          // BATCH*SIGP floats
    float* w_pad = sig + (size_t)BATCH * SIGP;            // NOUT*SIGP floats

    pad_wout_kernel<<<(NOUT * SIGP + 255) / 256, 256, 0, stream>>>(w_out, w_pad);
    sig_path_kernel<<<BATCH, 512, 0, stream>>>(x, w1, b1, w2, b2, w3, b3, sig);
    sig_gemm_out_kernel<<<BATCH / 16, 256, 0, stream>>>(sig, w_pad, b_out, out);
}